// TemporalGNN_49546742727295
// MI455X (gfx1250) — compile-verified
//
#include <hip/hip_runtime.h>

typedef __attribute__((ext_vector_type(2))) float v2f;
typedef __attribute__((ext_vector_type(8))) float v8f;

#define F_NODE 8
#define T_IN   8
#define T_OUT  4
#define HID    16

// ---- const-block layout (floats) ----
#define OFF_WZP  0      // 8x16 fused Wz@LzTop
#define OFF_WRP  128
#define OFF_WHP  256
#define OFF_LZB  384    // 16x16  Lz_w[16:32]
#define OFF_LRB  640
#define OFF_LHB  896
#define OFF_WOUT 1152   // 16x16  W_out padded to 16 cols
#define OFF_BZ   1408   // 16     bz@LzTop + Lz_b
#define OFF_BR   1424
#define OFF_BH   1440
#define OFF_BOUT 1456   // 16 (padded)
#define OFF_P    1472   // 8 softmax(attention)
#define CB_SIZE  2048

// ---------------- small helpers ----------------
__device__ __forceinline__ float fast_rcp(float x) {
  return __builtin_amdgcn_rcpf(x);  // single v_rcp_f32, no IEEE div fixup chain
}
__device__ __forceinline__ v8f vsplat(float x) {
  v8f r;
#pragma unroll
  for (int j = 0; j < 8; ++j) r[j] = x;
  return r;
}
__device__ __forceinline__ v8f vsigmoid(v8f x) {
  v8f r;
#pragma unroll
  for (int j = 0; j < 8; ++j) r[j] = fast_rcp(1.0f + __expf(-x[j]));
  return r;
}
__device__ __forceinline__ v8f vtanh(v8f x) {
  v8f r;
#pragma unroll
  for (int j = 0; j < 8; ++j) {
    float e = __expf(-2.0f * x[j]);
    r[j] = (1.0f - e) * fast_rcp(1.0f + e);
  }
  return r;
}

// chained fp32 WMMA: C += A(16xK) * B(Kx16), K = 4*NC
template <int NC>
__device__ __forceinline__ v8f mmacc(const float* a, const float* b, v8f c) {
#pragma unroll
  for (int i = 0; i < NC; ++i) {
    v2f av = {a[2 * i], a[2 * i + 1]};
    v2f bv = {b[2 * i], b[2 * i + 1]};
    c = __builtin_amdgcn_wmma_f32_16x16x4_f32(false, av, false, bv, (short)0, c,
                                              false, false);
  }
  return c;
}

// load a Kx16 row-major B matrix from global into B-fragment regs
// (VGPR0 of chunk c holds row 4c (+2 for hi half), VGPR1 the next row)
template <int NC>
__device__ __forceinline__ void loadB(const float* __restrict__ B, int col,
                                      int hi2, float* b) {
#pragma unroll
  for (int c = 0; c < NC; ++c) {
    int k = 4 * c + hi2;
    b[2 * c]     = B[k * 16 + col];
    b[2 * c + 1] = B[(k + 1) * 16 + col];
  }
}

#define LSTRIDE 18  // row stride (dwords): conflict-free, 8B aligned
__device__ __forceinline__ void c_to_lds(float* lptr, v8f c, int col, int hi) {
#pragma unroll
  for (int j = 0; j < 8; ++j) lptr[(j + 8 * hi) * LSTRIDE + col] = c[j];
}
__device__ __forceinline__ void lds_to_a(const float* lptr, float* a, int row,
                                         int hi2) {
#pragma unroll
  for (int c = 0; c < 4; ++c) {
    a[2 * c]     = lptr[row * LSTRIDE + 4 * c + hi2];
    a[2 * c + 1] = lptr[row * LSTRIDE + 4 * c + hi2 + 1];
  }
}

// ---------------- prep kernels ----------------
__global__ void k_zero(float* __restrict__ p, size_t n) {
  size_t i = (size_t)blockIdx.x * blockDim.x + threadIdx.x;
  size_t step = (size_t)gridDim.x * blockDim.x;
  for (; i < n; i += step) p[i] = 0.0f;
}

__global__ void k_deg(const int* __restrict__ ei, float* __restrict__ deg,
                      int E) {
  int e = blockIdx.x * blockDim.x + threadIdx.x;
  if (e >= E) return;
  atomicAdd(&deg[ei[E + e]], 1.0f);
}

__global__ void k_dinv(float* __restrict__ d, int N) {
  int i = blockIdx.x * blockDim.x + threadIdx.x;
  if (i >= N) return;
  d[i] = rsqrtf(d[i] + 1.0f);
}

// fuse weights: W' = W @ L_top, b' = b @ L_top + Lb ; copy L_bot; pad W_out
__global__ void k_prep(const float* att, const float* Wz, const float* bz,
                       const float* Lzw, const float* Lzb, const float* Wr,
                       const float* br, const float* Lrw, const float* Lrb,
                       const float* Wh, const float* bh, const float* Lhw,
                       const float* Lhb, const float* Wout, const float* bout,
                       float* __restrict__ CB) {
  int tid = threadIdx.x;
  for (int idx = tid; idx < 3 * 128; idx += 256) {  // W' matrices
    int m = idx / 128, r = idx % 128, f = r / 16, j = r % 16;
    const float* W  = (m == 0) ? Wz : (m == 1) ? Wr : Wh;
    const float* Lw = (m == 0) ? Lzw : (m == 1) ? Lrw : Lhw;
    float acc = 0.0f;
    for (int k = 0; k < 16; ++k) acc += W[f * 16 + k] * Lw[k * 16 + j];
    CB[OFF_WZP + m * 128 + r] = acc;
  }
  for (int idx = tid; idx < 3 * 256; idx += 256) {  // L_bot copies
    int m = idx / 256, r = idx % 256, k = r / 16, j = r % 16;
    const float* Lw = (m == 0) ? Lzw : (m == 1) ? Lrw : Lhw;
    CB[OFF_LZB + m * 256 + r] = Lw[(16 + k) * 16 + j];
  }
  if (tid < 256) {  // W_out padded 16x16
    int k = tid / 16, f = tid % 16;
    CB[OFF_WOUT + tid] = (f < 8) ? Wout[k * 8 + f] : 0.0f;
  }
  if (tid < 3 * 16) {  // fused biases
    int m = tid / 16, j = tid % 16;
    const float* b  = (m == 0) ? bz : (m == 1) ? br : bh;
    const float* Lw = (m == 0) ? Lzw : (m == 1) ? Lrw : Lhw;
    const float* Lb = (m == 0) ? Lzb : (m == 1) ? Lrb : Lhb;
    float acc = Lb[j];
    for (int k = 0; k < 16; ++k) acc += b[k] * Lw[k * 16 + j];
    CB[OFF_BZ + tid] = acc;
  }
  if (tid < 16) CB[OFF_BOUT + tid] = (tid < 8) ? bout[tid] : 0.0f;
  if (tid < 8) {  // softmax(attention)
    float m = att[0];
    for (int i = 1; i < 8; ++i) m = fmaxf(m, att[i]);
    float d = 0.0f;
    for (int i = 0; i < 8; ++i) d += __expf(att[i] - m);
    CB[OFF_P + tid] = __expf(att[tid] - m) * fast_rcp(d);
  }
}

// ---------------- edge aggregation ----------------
// acc[dst][f] += dinv[src]*dinv[dst] * X[src][f]   (one thread per edge)
__global__ void k_scatter(const int* __restrict__ ei,
                          const float* __restrict__ dinv,
                          const float* __restrict__ xsrc, int fstride,
                          int nstride, float* __restrict__ acc, int E) {
  int e = blockIdx.x * blockDim.x + threadIdx.x;
  if (e >= E) return;
  int s = ei[e], d = ei[E + e];
  float w = dinv[s] * dinv[d];
  const float* xp = xsrc + (size_t)s * nstride;
#pragma unroll
  for (int f = 0; f < 8; ++f)
    atomicAdd(&acc[(size_t)d * 8 + f], w * xp[f * fstride]);
}

// acc[i][f] += dinv[i]^2 * X[i][f]  (self term -> Xeff complete)
__global__ void k_finish(const float* __restrict__ dinv,
                         const float* __restrict__ xsrc, int fstride,
                         int nstride, float* __restrict__ acc, int N) {
  int i = blockIdx.x * blockDim.x + threadIdx.x;
  if (i >= N) return;
  float w = dinv[i] * dinv[i];
  const float* xp = xsrc + (size_t)i * nstride;
#pragma unroll
  for (int f = 0; f < 8; ++f) acc[(size_t)i * 8 + f] += w * xp[f * fstride];
}

// ---------------- fused GRU sweep: one 16-node tile per wave32 ----------------
__global__ void __launch_bounds__(256) k_tgcn(
    const float* __restrict__ CB, const float* __restrict__ XeffO,
    const float* __restrict__ XeffP, float* __restrict__ Pred,
    float* __restrict__ out, int N, int s) {
  __shared__ float lds[8 * 16 * LSTRIDE];
  const int lane = threadIdx.x & 31;
  const int wave = threadIdx.x >> 5;
  const int col  = lane & 15;   // C-layout: N index / B-layout column
  const int hi   = lane >> 4;
  const int hi2  = hi * 2;
  const int row  = col;         // A-layout: M index
  float* lptr = lds + wave * (16 * LSTRIDE);

  int ntiles = (N + 15) / 16;
  int tile = blockIdx.x * 8 + wave;
  bool valid = tile < ntiles;
  if (!valid) tile = 0;
  const int node0 = tile * 16;

  // cache all B fragments (weights) in registers
  float wzp[4], wrp[4], whp[4], lzb[8], lrb[8], lhb[8];
  loadB<2>(CB + OFF_WZP, col, hi2, wzp);
  loadB<2>(CB + OFF_WRP, col, hi2, wrp);
  loadB<2>(CB + OFF_WHP, col, hi2, whp);
  loadB<4>(CB + OFF_LZB, col, hi2, lzb);
  loadB<4>(CB + OFF_LRB, col, hi2, lrb);
  loadB<4>(CB + OFF_LHB, col, hi2, lhb);
  const float bzv = CB[OFF_BZ + col];
  const float brv = CB[OFF_BR + col];
  const float bhv = CB[OFF_BH + col];

  float ha[8];  // H in A-fragment layout
#pragma unroll
  for (int i = 0; i < 8; ++i) ha[i] = 0.0f;
  v8f hc = vsplat(0.0f);    // H in C layout
  v8f hacc = vsplat(0.0f);  // attention-weighted accumulation

  for (int t = 0; t < T_IN; ++t) {
    const float* Xcol = (t < T_IN - s)
                            ? XeffO + (size_t)(s + t) * N * 8
                            : XeffP + (size_t)(t - T_IN + s) * N * 8;
    // Xeff tile as 16x8 A fragment (2 chunks of 16x4)
    const float* xr = Xcol + (size_t)(node0 + row) * 8;
    float xa[4];
    xa[0] = xr[hi2];     xa[1] = xr[hi2 + 1];
    xa[2] = xr[4 + hi2]; xa[3] = xr[5 + hi2];

    // Z / R gates
    v8f zc = vsplat(bzv);
    zc = mmacc<2>(xa, wzp, zc);
    zc = mmacc<4>(ha, lzb, zc);
    v8f rc = vsplat(brv);
    rc = mmacc<2>(xa, wrp, rc);
    rc = mmacc<4>(ha, lrb, rc);
    zc = vsigmoid(zc);
    rc = vsigmoid(rc);

    // H*R  ->  A layout via LDS transpose
    v8f hrc = hc * rc;
    __syncthreads();
    c_to_lds(lptr, hrc, col, hi);
    __syncthreads();
    float hra[8];
    lds_to_a(lptr, hra, row, hi2);

    // candidate state
    v8f tc = vsplat(bhv);
    tc = mmacc<2>(xa, whp, tc);
    tc = mmacc<4>(hra, lhb, tc);
    tc = vtanh(tc);

    hc = zc * hc + (1.0f - zc) * tc;
    float p = CB[OFF_P + t];
    hacc += p * hc;

    // new H -> A layout for next step
    __syncthreads();
    c_to_lds(lptr, hc, col, hi);
    __syncthreads();
    lds_to_a(lptr, ha, row, hi2);
  }

  // output projection: relu(H_acc) @ W_out + b_out
#pragma unroll
  for (int j = 0; j < 8; ++j) hacc[j] = fmaxf(hacc[j], 0.0f);
  __syncthreads();
  c_to_lds(lptr, hacc, col, hi);
  __syncthreads();
  float aa[8];
  lds_to_a(lptr, aa, row, hi2);
  float wout[8];
  loadB<4>(CB + OFF_WOUT, col, hi2, wout);
  v8f oc = vsplat(CB[OFF_BOUT + col]);
  oc = mmacc<4>(aa, wout, oc);

  if (valid && col < F_NODE) {
#pragma unroll
    for (int j = 0; j < 8; ++j) {
      int node = node0 + j + 8 * hi;
      if (node < N) {
        out[(size_t)node * (F_NODE * T_OUT) + col * T_OUT + s] = oc[j];
        if (s < T_OUT - 1)
          Pred[(size_t)s * N * 8 + (size_t)node * 8 + col] = oc[j];
      }
    }
  }
}

// ---------------- host driver ----------------
extern "C" void kernel_launch(void* const* d_in, const int* in_sizes, int n_in,
                              void* d_out, int out_size, void* d_ws,
                              size_t ws_size, hipStream_t stream) {
  const float* x    = (const float*)d_in[0];
  const int*   ei   = (const int*)d_in[1];
  const float* att  = (const float*)d_in[2];
  const float* Wz   = (const float*)d_in[3];
  const float* bz   = (const float*)d_in[4];
  const float* Lzw  = (const float*)d_in[5];
  const float* Lzb  = (const float*)d_in[6];
  const float* Wr   = (const float*)d_in[7];
  const float* br   = (const float*)d_in[8];
  const float* Lrw  = (const float*)d_in[9];
  const float* Lrb  = (const float*)d_in[10];
  const float* Wh   = (const float*)d_in[11];
  const float* bh   = (const float*)d_in[12];
  const float* Lhw  = (const float*)d_in[13];
  const float* Lhb  = (const float*)d_in[14];
  const float* Wout = (const float*)d_in[15];
  const float* bout = (const float*)d_in[16];

  const int N = in_sizes[0] / (F_NODE * T_IN);
  const int E = in_sizes[1] / 2;

  float* ws    = (float*)d_ws;
  float* dinv  = ws;                          // N
  float* CB    = ws + N;                      // CB_SIZE
  float* XeffO = CB + CB_SIZE;                // 8 * N * 8
  float* XeffP = XeffO + (size_t)T_IN * N * 8;  // 3 * N * 8
  float* Pred  = XeffP + (size_t)3 * N * 8;     // 3 * N * 8

  // zero: dinv + CB + XeffO + XeffP (Pred fully written before read)
  size_t zcount = (size_t)N + CB_SIZE + (size_t)(T_IN + 3) * N * 8;
  k_zero<<<2048, 256, 0, stream>>>(ws, zcount);

  k_deg<<<(E + 255) / 256, 256, 0, stream>>>(ei, dinv, E);
  k_dinv<<<(N + 255) / 256, 256, 0, stream>>>(dinv, N);
  k_prep<<<1, 256, 0, stream>>>(att, Wz, bz, Lzw, Lzb, Wr, br, Lrw, Lrb, Wh, bh,
                                Lhw, Lhb, Wout, bout, CB);

  // aggregate the 8 original feature columns once
  for (int t = 0; t < T_IN; ++t) {
    float* xc = XeffO + (size_t)t * N * 8;
    k_scatter<<<(E + 255) / 256, 256, 0, stream>>>(ei, dinv, x + t, T_IN,
                                                   F_NODE * T_IN, xc, E);
    k_finish<<<(N + 255) / 256, 256, 0, stream>>>(dinv, x + t, T_IN,
                                                  F_NODE * T_IN, xc, N);
  }

  const int ntiles = (N + 15) / 16;
  const int tgrid = (ntiles + 7) / 8;
  for (int s = 0; s < T_OUT; ++s) {
    k_tgcn<<<tgrid, 256, 0, stream>>>(CB, XeffO, XeffP, Pred, (float*)d_out, N,
                                      s);
    if (s < T_OUT - 1) {  // aggregate the freshly predicted column
      float* pc = Pred + (size_t)s * N * 8;
      float* xc = XeffP + (size_t)s * N * 8;
      k_scatter<<<(E + 255) / 256, 256, 0, stream>>>(ei, dinv, pc, 1, 8, xc, E);
      k_finish<<<(N + 255) / 256, 256, 0, stream>>>(dinv, pc, 1, 8, xc, N);
    }
  }
}